// ParameterizedQuantumCircuit_81312320848294
// MI455X (gfx1250) — compile-verified
//
#include <hip/hip_runtime.h>
#include <math.h>

#define NQ 10
#define NL 5
#define QDIM 1024

typedef __attribute__((ext_vector_type(2))) float v2f;
typedef __attribute__((ext_vector_type(8))) float v8f;

// d_ws float layout:
//   URr[6][256]  @ 0       (U_row = kron(u0..u3), real)
//   URi[6][256]  @ 1536
//   WTr[6][4096] @ 3072    (WT[k][c] = W[c][k], W = kron(u4..u9), real)
//   WTi[6][4096] @ 27648
#define OFF_URR 0
#define OFF_URI 1536
#define OFF_WTR 3072
#define OFF_WTI 27648

struct cplx { float re, im; };
__device__ __forceinline__ cplx cmul(cplx a, cplx b) {
  return { a.re * b.re - a.im * b.im, a.re * b.im + a.im * b.re };
}

// Build per-layer group unitaries. One block per layer (6 blocks x 256 threads).
__global__ void pqc_build_gates(const float* __restrict__ theta,
                                float* __restrict__ ws) {
  const int l   = blockIdx.x;     // 0..5
  const int tid = threadIdx.x;    // 0..255
  __shared__ cplx u[NQ][2][2];

  if (tid < NQ) {
    const int q = tid;
    const float t0 = theta[(l * NQ + q) * 3 + 0];
    const float t1 = theta[(l * NQ + q) * 3 + 1];
    const float t2 = theta[(l * NQ + q) * 3 + 2];
    const float c0 = cosf(0.5f * t0), s0 = sinf(0.5f * t0);
    const float c1 = cosf(0.5f * t1), s1 = sinf(0.5f * t1);
    const float c2 = cosf(0.5f * t2), s2 = sinf(0.5f * t2);
    // M = Ry(t1) @ Rx(t0)
    const cplx M00 = {  c1 * c0,  s1 * s0 };
    const cplx M01 = { -s1 * c0, -c1 * s0 };
    const cplx M10 = {  s1 * c0, -c1 * s0 };
    const cplx M11 = {  c1 * c0, -s1 * s0 };
    const cplx e0  = { c2, -s2 };  // exp(-i t2/2)
    const cplx e1  = { c2,  s2 };  // exp(+i t2/2)
    u[q][0][0] = cmul(e0, M00);
    u[q][0][1] = cmul(e0, M01);
    u[q][1][0] = cmul(e1, M10);
    u[q][1][1] = cmul(e1, M11);
  }
  __syncthreads();

  // U_row = kron(u0,u1,u2,u3): 16x16. qubit0 <-> MSB of row index.
  {
    const int r = tid >> 4, c = tid & 15;
    cplx acc = { 1.f, 0.f };
#pragma unroll
    for (int q = 0; q < 4; ++q) {
      const int sh = 3 - q;
      acc = cmul(acc, u[q][(r >> sh) & 1][(c >> sh) & 1]);
    }
    ws[OFF_URR + l * 256 + tid] = acc.re;
    ws[OFF_URI + l * 256 + tid] = acc.im;
  }
  // WT[k][c] = W[c][k], W = kron(u4..u9): 64x64. qubit4 <-> bit5.
  for (int idx = tid; idx < 4096; idx += 256) {
    const int k = idx >> 6, c = idx & 63;
    cplx acc = { 1.f, 0.f };
#pragma unroll
    for (int q = 4; q < 10; ++q) {
      const int sh = 9 - q;
      acc = cmul(acc, u[q][(c >> sh) & 1][(k >> sh) & 1]);
    }
    ws[OFF_WTR + l * 4096 + idx] = acc.re;
    ws[OFF_WTI + l * 4096 + idx] = acc.im;
  }
}

// One wave32 per batch element; state in LDS; variational layers via WMMA f32.
__global__ __launch_bounds__(32)
void pqc_simulate(const float* __restrict__ x, const float* __restrict__ lam,
                  const float* __restrict__ w, const float* __restrict__ ws,
                  float* __restrict__ out) {
  const int b    = blockIdx.x;
  const int lane = threadIdx.x;
  const int hi   = lane >> 4;    // 0 or 1
  const int lo   = lane & 15;
  __shared__ float Sr[QDIM], Si[QDIM], Tr[QDIM], Ti[QDIM];

  // |0...0>
  for (int i = lane; i < QDIM; i += 32) { Sr[i] = 0.f; Si[i] = 0.f; }
  if (lane == 0) Sr[0] = 1.f;
  __syncthreads();

  float xv[NQ];
#pragma unroll
  for (int q = 0; q < NQ; ++q) xv[q] = x[b * NQ + q];

  for (int l = 0; l <= NL; ++l) {
    const float* URr = ws + OFF_URR + l * 256;
    const float* URi = ws + OFF_URI + l * 256;
    const float* WTr = ws + OFF_WTR + l * 4096;
    const float* WTi = ws + OFF_WTI + l * 4096;

    // ---- row op: T(16x64) = U_row(16x16) * S(16x64) ----
    for (int ct = 0; ct < 4; ++ct) {
      const int c0 = ct * 16 + lo;
      v8f accR = { 0.f, 0.f, 0.f, 0.f, 0.f, 0.f, 0.f, 0.f };
      v8f accI = { 0.f, 0.f, 0.f, 0.f, 0.f, 0.f, 0.f, 0.f };
#pragma unroll
      for (int kk = 0; kk < 4; ++kk) {
        const int kb = kk * 4 + 2 * hi;
        // A fragment (16x4) of U_row: m=lo, k=kb+j (contiguous -> b64 load)
        const float2 arp = *(const float2*)(URr + lo * 16 + kb);
        const float2 aip = *(const float2*)(URi + lo * 16 + kb);
        const v2f Ar  = {  arp.x,  arp.y };
        const v2f Ai  = {  aip.x,  aip.y };
        const v2f mAi = { -aip.x, -aip.y };
        // B fragment (4x16) of S: row k=kb+j, col c0
        const v2f Br = { Sr[kb * 64 + c0], Sr[(kb + 1) * 64 + c0] };
        const v2f Bi = { Si[kb * 64 + c0], Si[(kb + 1) * 64 + c0] };
        accR = __builtin_amdgcn_wmma_f32_16x16x4_f32(false, Ar,  false, Br, (short)0, accR, false, false);
        accR = __builtin_amdgcn_wmma_f32_16x16x4_f32(false, mAi, false, Bi, (short)0, accR, false, false);
        accI = __builtin_amdgcn_wmma_f32_16x16x4_f32(false, Ar,  false, Bi, (short)0, accI, false, false);
        accI = __builtin_amdgcn_wmma_f32_16x16x4_f32(false, Ai,  false, Br, (short)0, accI, false, false);
      }
#pragma unroll
      for (int v = 0; v < 8; ++v) {
        const int M = v + 8 * hi;
        Tr[M * 64 + c0] = accR[v];
        Ti[M * 64 + c0] = accI[v];
      }
    }
    __syncthreads();

    // ---- col op: S(16x64) = T(16x64) * W^T(64x64) ----
    for (int ct = 0; ct < 4; ++ct) {
      const int c0 = ct * 16 + lo;
      v8f accR = { 0.f, 0.f, 0.f, 0.f, 0.f, 0.f, 0.f, 0.f };
      v8f accI = { 0.f, 0.f, 0.f, 0.f, 0.f, 0.f, 0.f, 0.f };
      for (int kk = 0; kk < 16; ++kk) {
        const int kb = kk * 4 + 2 * hi;
        // A fragment (16x4) of T: m=lo, k=kb+j (contiguous -> ds b64)
        const v2f Ar  = { Tr[lo * 64 + kb], Tr[lo * 64 + kb + 1] };
        const v2f Ai  = { Ti[lo * 64 + kb], Ti[lo * 64 + kb + 1] };
        const v2f mAi = -Ai;
        // B fragment (4x16) of W^T: row k=kb+j, col c0
        const v2f Bwr = { WTr[kb * 64 + c0], WTr[(kb + 1) * 64 + c0] };
        const v2f Bwi = { WTi[kb * 64 + c0], WTi[(kb + 1) * 64 + c0] };
        accR = __builtin_amdgcn_wmma_f32_16x16x4_f32(false, Ar,  false, Bwr, (short)0, accR, false, false);
        accR = __builtin_amdgcn_wmma_f32_16x16x4_f32(false, mAi, false, Bwi, (short)0, accR, false, false);
        accI = __builtin_amdgcn_wmma_f32_16x16x4_f32(false, Ar,  false, Bwi, (short)0, accI, false, false);
        accI = __builtin_amdgcn_wmma_f32_16x16x4_f32(false, Ai,  false, Bwr, (short)0, accI, false, false);
      }
#pragma unroll
      for (int v = 0; v < 8; ++v) {
        const int M = v + 8 * hi;
        Sr[M * 64 + c0] = accR[v];
        Si[M * 64 + c0] = accI[v];
      }
    }
    __syncthreads();

    // ---- CZ ring diagonal: sign = (-1)^popcount(k & rot1(k)) ----
    for (int i = lane; i < QDIM; i += 32) {
      const int j = ((i << 1) & 1023) | (i >> 9);
      if (__popc(i & j) & 1) { Sr[i] = -Sr[i]; Si[i] = -Si[i]; }
    }
    __syncthreads();

    // ---- data-encoding Rx(x*lambda) on every qubit (layers 0..4) ----
    if (l < NL) {
      for (int q = 0; q < NQ; ++q) {
        const int p = 9 - q;
        const int h = 1 << p;
        const float ang = xv[q] * lam[l * NQ + q];
        const float ch = cosf(0.5f * ang), sn = sinf(0.5f * ang);
        for (int i = 0; i < 16; ++i) {
          const int t  = i * 32 + lane;             // pair index 0..511
          const int k0 = ((t >> p) << (p + 1)) | (t & (h - 1));
          const int k1 = k0 | h;
          const float ar = Sr[k0], ai = Si[k0];
          const float br = Sr[k1], bi = Si[k1];
          Sr[k0] = ch * ar + sn * bi;
          Si[k0] = ch * ai - sn * br;
          Sr[k1] = ch * br + sn * ai;
          Si[k1] = ch * bi - sn * ar;
        }
        __syncthreads();
      }
    }
  }

  // ---- <Z x ... x Z> and 2-way softmax ----
  float local = 0.f;
  for (int i = lane; i < QDIM; i += 32) {
    const float p2 = Sr[i] * Sr[i] + Si[i] * Si[i];
    local += (__popc(i) & 1) ? -p2 : p2;
  }
  Tr[lane] = local;
  __syncthreads();
  if (lane == 0) {
    float e = 0.f;
    for (int i = 0; i < 32; ++i) e += Tr[i];
    const float l0 = e * w[0], l1 = e * w[1];   // BETA = 1
    const float m  = fmaxf(l0, l1);
    const float e0 = expf(l0 - m), e1 = expf(l1 - m);
    const float inv = 1.f / (e0 + e1);
    out[b * 2 + 0] = e0 * inv;
    out[b * 2 + 1] = e1 * inv;
  }
}

extern "C" void kernel_launch(void* const* d_in, const int* in_sizes, int n_in,
                              void* d_out, int out_size, void* d_ws, size_t ws_size,
                              hipStream_t stream) {
  (void)n_in; (void)out_size; (void)ws_size;
  const float* x     = (const float*)d_in[0];
  const float* theta = (const float*)d_in[1];
  const float* lam   = (const float*)d_in[2];
  const float* w     = (const float*)d_in[3];
  float* out = (float*)d_out;
  float* ws  = (float*)d_ws;
  const int B = in_sizes[0] / NQ;   // 2048

  pqc_build_gates<<<dim3(NL + 1), dim3(256), 0, stream>>>(theta, ws);
  pqc_simulate<<<dim3(B), dim3(32), 0, stream>>>(x, lam, w, ws, out);
}